// SketchDecoder_52295521796332
// MI455X (gfx1250) — compile-verified
//
#include <hip/hip_runtime.h>
#include <stdint.h>

#define HIDDEN  4096
#define STEPS   1024
#define THREADS 512
#define WAVES   (THREADS / 32)   // 16 waves -> 4 waves per SIMD32
#define JPW     (HIDDEN / WAVES) // 256 h-values per wave
#define CHUNKS  (JPW / 16)       // 16 x 16-row WMMA chunks per wave
#define NCHUNKS (HIDDEN / 16)    // 256 chunks total across the block

#define WS_STRIDE   16                                        // floats per j (64 B)
#define TBL_FLOATS  ((size_t)HIDDEN * WS_STRIDE)              // 65536 floats (256 KB)
#define AWS_FLOATS  ((size_t)3 * NCHUNKS * 32 * 2)            // 49152 floats (192 KB)
#define WS_BYTES    ((TBL_FLOATS + AWS_FLOATS) * sizeof(float))  // 448 KB

typedef __attribute__((ext_vector_type(2))) float v2f;
typedef __attribute__((ext_vector_type(4))) float v4f;
typedef __attribute__((ext_vector_type(8))) float v8f;

// Global-address-space pointer types: loads through these lower to global_load_*
// (LOADcnt only), not flat_load_* (LOADcnt + DScnt + aperture probe).
typedef __attribute__((address_space(1))) const float* gcf;
typedef __attribute__((address_space(1))) const v2f*   gcv2;
typedef __attribute__((address_space(1))) const v4f*   gcv4;

__device__ __forceinline__ float rcp_fast(float x) { return __builtin_amdgcn_rcpf(x); }
__device__ __forceinline__ float exp_fast(float x) { return __expf(x); }
__device__ __forceinline__ float sigmoid_f(float x) { return rcp_fast(1.0f + exp_fast(-x)); }
__device__ __forceinline__ float tanh_f(float x) {
    float xc = fminf(fmaxf(x, -15.0f), 15.0f);
    float e  = exp_fast(2.0f * xc);
    return (e - 1.0f) * rcp_fast(e + 1.0f);
}
// Pointer laundering: produces a fresh "unknown" SGPR value so derived loads are
// re-issued from cache (L0-hot) each step instead of being hoisted by LICM into
// hundreds of live VGPRs and spilled to scratch. Returns an addrspace(1) pointer
// so the loads still lower to global_load_* rather than flat_load_*.
__device__ __forceinline__ gcf launder_gptr(const float* p) {
    uintptr_t a = (uintptr_t)p;
    asm volatile("" : "+s"(a));
    return (gcf)a;
}
// Constant-index select tree: keeps the v8f accumulator in VGPRs (no scratch).
__device__ __forceinline__ float extract8(v8f d, int r) {
    float x = d[0];
    x = (r == 1) ? d[1] : x;
    x = (r == 2) ? d[2] : x;
    x = (r == 3) ? d[3] : x;
    x = (r == 4) ? d[4] : x;
    x = (r == 5) ? d[5] : x;
    x = (r == 6) ? d[6] : x;
    x = (r == 7) ? d[7] : x;
    return x;
}

// ---- Prep kernel 1: per-j 64-byte record.
//   [0..6]  = W_lin[o][j] transposed
//   [8,9]   = W_ih[j,4],       (b_ih+b_hh)[j]         (i-gate tail)
//   [10,11] = W_ih[8192+j,4],  (b_ih+b_hh)[8192+j]    (g-gate tail)
//   [12,13] = W_ih[12288+j,4], (b_ih+b_hh)[12288+j]   (o-gate tail)
__global__ void pack_tables_kernel(const float* __restrict__ W_ih,
                                   const float* __restrict__ b_ih,
                                   const float* __restrict__ b_hh,
                                   const float* __restrict__ W_lin,
                                   float* __restrict__ ws)
{
    int j = blockIdx.x * blockDim.x + threadIdx.x;
    if (j >= HIDDEN) return;
    float* t = ws + (size_t)j * WS_STRIDE;
#pragma unroll
    for (int o = 0; o < 7; ++o) t[o] = W_lin[o * HIDDEN + j];
    t[7]  = 0.f;
    t[8]  = W_ih[j * 5 + 4];             t[9]  = b_ih[j] + b_hh[j];
    t[10] = W_ih[(8192 + j) * 5 + 4];    t[11] = b_ih[8192 + j] + b_hh[8192 + j];
    t[12] = W_ih[(12288 + j) * 5 + 4];   t[13] = b_ih[12288 + j] + b_hh[12288 + j];
    t[14] = 0.f;  t[15] = 0.f;
}

// ---- Prep kernel 2: WMMA A tiles (W_ih columns 0..3) pre-swizzled into the exact
// per-lane VGPR layout of V_WMMA_F32_16X16X4_F32 A operands:
//   lanes 0-15 hold (K=0,K=1), lanes 16-31 hold (K=2,K=3), M = lane%16.
// One coalesced global_load_b64 per gate per chunk in the hot loop.
__global__ void pack_awma_kernel(const float* __restrict__ W_ih,
                                 float* __restrict__ aws)
{
    int t = blockIdx.x * blockDim.x + threadIdx.x;   // (cg, lane) pair
    if (t >= NCHUNKS * 32) return;
    const int cg   = t >> 5;
    const int lane = t & 31;
    const int m    = lane & 15;
    const int ko   = (lane < 16) ? 0 : 2;
    const int jb   = cg * 16;
    const int rI = (jb + m) * 5;
    const int rG = (8192  + jb + m) * 5;
    const int rO = (12288 + jb + m) * 5;
    v2f* a = (v2f*)aws;
    a[(0 * NCHUNKS + cg) * 32 + lane] = v2f{W_ih[rI + ko], W_ih[rI + ko + 1]};
    a[(1 * NCHUNKS + cg) * 32 + lane] = v2f{W_ih[rG + ko], W_ih[rG + ko + 1]};
    a[(2 * NCHUNKS + cg) * 32 + lane] = v2f{W_ih[rO + ko], W_ih[rO + ko + 1]};
}

template <bool UseWs>
__global__ __launch_bounds__(THREADS)
void sketch_decoder_kernel(const float* __restrict__ W_ih,
                           const float* __restrict__ b_ih,
                           const float* __restrict__ b_hh,
                           const float* __restrict__ W_lin,
                           const float* __restrict__ b_lin,
                           const float* __restrict__ eps,
                           const float* __restrict__ ws,
                           float* __restrict__ out)
{
    __shared__ float s_red[WAVES][8];
    __shared__ float s_prev[8];
    __shared__ int   s_done;

    const int tid  = threadIdx.x;
    const int wid  = tid >> 5;
    const int lane = tid & 31;
    const int mrow = lane & 15;                        // A-matrix M row this lane holds
    const int ko   = (lane < 16) ? 0 : 2;              // A/B K offset per half-wave
    const int rsel = lane & 7;                         // D register row this lane extracts
    const int msel = (lane & 7) + ((lane >> 4) << 3);  // matrix row 0..15 this lane owns

    if (tid == 0) {
        s_prev[0] = 0.f; s_prev[1] = 0.f; s_prev[2] = 1.f; s_prev[3] = 0.f; s_prev[4] = 0.f;
        s_done = 0;
    }
    __syncthreads();

    int stop_step = STEPS;
    for (int s = 0; s < STEPS; ++s) {
        // Broadcast the recurrent 5-vector; build WMMA B operand (B[k,n] = prev[k]).
        const float p0 = s_prev[0], p1 = s_prev[1], p2 = s_prev[2],
                    p3 = s_prev[3], p4 = s_prev[4];
        v2f b;
        b.x = (lane < 16) ? p0 : p2;
        b.y = (lane < 16) ? p1 : p3;

        float part[7] = {0.f, 0.f, 0.f, 0.f, 0.f, 0.f, 0.f};
        const v8f zero = {0.f, 0.f, 0.f, 0.f, 0.f, 0.f, 0.f, 0.f};

#pragma unroll
        for (int c = 0; c < CHUNKS; ++c) {
            const int cg = wid * CHUNKS + c;           // global chunk id
            const int jb = cg * 16;
            const int j  = jb + msel;  // h index this lane owns (each j held by 2 lanes)

            v2f ai, ag, ao;
            float wl0, wl1, wl2, wl3, wl4, wl5, wl6;
            float w4i, bsi, w4g, bsg, w4o, bso;
            if (UseWs) {
                // Fresh laundered base per chunk: loads stay inside the step loop
                // (L0-hot) and at most ~1 chunk of loads is in flight at once.
                gcf  wsl  = launder_gptr(ws);
                gcv2 awsl = (gcv2)(wsl + TBL_FLOATS);
                ai = awsl[(0 * NCHUNKS + cg) * 32 + lane];
                ag = awsl[(1 * NCHUNKS + cg) * 32 + lane];
                ao = awsl[(2 * NCHUNKS + cg) * 32 + lane];
                gcv4 tp = (gcv4)(wsl + (size_t)j * WS_STRIDE);
                v4f qa = tp[0], qb = tp[1], qc = tp[2], qd = tp[3];
                wl0 = qa.x; wl1 = qa.y; wl2 = qa.z; wl3 = qa.w;
                wl4 = qb.x; wl5 = qb.y; wl6 = qb.z;
                w4i = qc.x; bsi = qc.y; w4g = qc.z; bsg = qc.w;
                w4o = qd.x; bso = qd.y;
            } else {
                gcf Wi = launder_gptr(W_ih);
                gcf Wl = launder_gptr(W_lin);
                gcf bi = launder_gptr(b_ih);
                gcf bh = launder_gptr(b_hh);
                const int rI = (jb + mrow) * 5;
                const int rG = (8192  + jb + mrow) * 5;
                const int rO = (12288 + jb + mrow) * 5;
                ai = v2f{Wi[rI + ko], Wi[rI + ko + 1]};
                ag = v2f{Wi[rG + ko], Wi[rG + ko + 1]};
                ao = v2f{Wi[rO + ko], Wi[rO + ko + 1]};
                wl0 = Wl[0 * HIDDEN + j]; wl1 = Wl[1 * HIDDEN + j];
                wl2 = Wl[2 * HIDDEN + j]; wl3 = Wl[3 * HIDDEN + j];
                wl4 = Wl[4 * HIDDEN + j]; wl5 = Wl[5 * HIDDEN + j];
                wl6 = Wl[6 * HIDDEN + j];
                w4i = Wi[j * 5 + 4];             bsi = bi[j] + bh[j];
                w4g = Wi[(8192 + j) * 5 + 4];    bsg = bi[8192 + j] + bh[8192 + j];
                w4o = Wi[(12288 + j) * 5 + 4];   bso = bi[12288 + j] + bh[12288 + j];
            }

            // gates(k=0..3) on the matrix pipe: D = W_ih[rows,0:4] x prev[0:4]
            v8f di = __builtin_amdgcn_wmma_f32_16x16x4_f32(false, ai, false, b,
                                                           (short)0, zero, false, false);
            v8f dg = __builtin_amdgcn_wmma_f32_16x16x4_f32(false, ag, false, b,
                                                           (short)0, zero, false, false);
            v8f dq = __builtin_amdgcn_wmma_f32_16x16x4_f32(false, ao, false, b,
                                                           (short)0, zero, false, false);

            // k=4 column + fused biases, then activations (f-gate term vanishes, c0==0).
            float gi = extract8(di, rsel) + w4i * p4 + bsi;
            float gg = extract8(dg, rsel) + w4g * p4 + bsg;
            float go = extract8(dq, rsel) + w4o * p4 + bso;
            float cc = sigmoid_f(gi) * tanh_f(gg);
            float h  = sigmoid_f(go) * tanh_f(cc);

            // All 32 lanes accumulate (each j counted exactly twice across the wave);
            // final reduction scales by 0.5 -> exact, no EXEC divergence in the loop.
            part[0] = fmaf(wl0, h, part[0]);
            part[1] = fmaf(wl1, h, part[1]);
            part[2] = fmaf(wl2, h, part[2]);
            part[3] = fmaf(wl3, h, part[3]);
            part[4] = fmaf(wl4, h, part[4]);
            part[5] = fmaf(wl5, h, part[5]);
            part[6] = fmaf(wl6, h, part[6]);
        }

        // Wave-level reduction of the 7 logit partials.
#pragma unroll
        for (int off = 16; off > 0; off >>= 1) {
#pragma unroll
            for (int o = 0; o < 7; ++o)
                part[o] += __shfl_down(part[o], off, 32);
        }
        if (lane == 0) {
#pragma unroll
            for (int o = 0; o < 7; ++o) s_red[wid][o] = part[o];
        }
        __syncthreads();

        // Wave 0: cross-wave reduction, logits, sampling, done logic (no extra barrier:
        // the 7 logits are exchanged with wave32 shuffles inside wave 0).
        if (wid == 0) {
            float acc = 0.f;
            float bl  = 0.f;
            if (lane < 7) {
#pragma unroll
                for (int w2 = 0; w2 < WAVES; ++w2) acc += s_red[w2][lane];
                bl = b_lin[lane];
            }
            float lg = tanh_f(acc * 0.5f + bl);     // 0.5: each j accumulated twice
            const float l0 = __shfl(lg, 0, 32), l1 = __shfl(lg, 1, 32),
                        l2 = __shfl(lg, 2, 32), l3 = __shfl(lg, 3, 32),
                        l4 = __shfl(lg, 4, 32), l5 = __shfl(lg, 5, 32),
                        l6 = __shfl(lg, 6, 32);
            if (lane == 0) {
                const float ex = eps[2 * s + 0], ey = eps[2 * s + 1];
                float sx = l0 + exp_fast(0.5f * l1) * ex;
                float sy = l2 + exp_fast(0.5f * l3) * ey;
                float mx = fmaxf(l4, fmaxf(l5, l6));
                float e4 = exp_fast(l4 - mx), e5 = exp_fast(l5 - mx), e6 = exp_fast(l6 - mx);
                float rs = rcp_fast(e4 + e5 + e6);
                // done uses the row that FED this step (argmax(prev[2:]) == 2).
                const float q2 = s_prev[2], q3 = s_prev[3], q4 = s_prev[4];
                int dn = s_done | (int)((q4 > q2) && (q4 > q3));
                float r0 = dn ? 0.f : sx;
                float r1 = dn ? 0.f : sy;
                float r2 = dn ? 0.f : e4 * rs;
                float r3 = dn ? 0.f : e5 * rs;
                float r4 = dn ? 0.f : e6 * rs;
                float* op = out + s * 5;
                op[0] = r0; op[1] = r1; op[2] = r2; op[3] = r3; op[4] = r4;
                s_prev[0] = r0; s_prev[1] = r1; s_prev[2] = r2; s_prev[3] = r3; s_prev[4] = r4;
                s_done = dn;
            }
        }
        __syncthreads();
        if (s_done) { stop_step = s + 1; break; }   // sticky done: all later rows are zero
    }

    // Bulk-zero every row after the stop flag fired (reference emits zeros there).
    for (int i = stop_step * 5 + tid; i < STEPS * 5; i += THREADS) out[i] = 0.f;
}

extern "C" void kernel_launch(void* const* d_in, const int* in_sizes, int n_in,
                              void* d_out, int out_size, void* d_ws, size_t ws_size,
                              hipStream_t stream) {
    const float* W_ih  = (const float*)d_in[0];
    // d_in[1] = W_hh: intentionally unused — the recurrence always sees h0 == 0,
    // so the 16384x4096 GEMV in the reference is algebraically zero.
    const float* b_ih  = (const float*)d_in[2];
    const float* b_hh  = (const float*)d_in[3];
    const float* W_lin = (const float*)d_in[4];
    const float* b_lin = (const float*)d_in[5];
    const float* eps   = (const float*)d_in[6];
    float* out = (float*)d_out;

    if (ws_size >= WS_BYTES) {
        float* ws = (float*)d_ws;
        pack_tables_kernel<<<dim3((HIDDEN + 255) / 256), dim3(256), 0, stream>>>(
            W_ih, b_ih, b_hh, W_lin, ws);
        pack_awma_kernel<<<dim3((NCHUNKS * 32 + 255) / 256), dim3(256), 0, stream>>>(
            W_ih, ws + TBL_FLOATS);
        sketch_decoder_kernel<true><<<dim3(1), dim3(THREADS), 0, stream>>>(
            W_ih, b_ih, b_hh, W_lin, b_lin, eps, ws, out);
    } else {
        sketch_decoder_kernel<false><<<dim3(1), dim3(THREADS), 0, stream>>>(
            W_ih, b_ih, b_hh, W_lin, b_lin, eps, (const float*)nullptr, out);
    }
}